// BasicEncoder_16475494548018
// MI455X (gfx1250) — compile-verified
//
#include <hip/hip_runtime.h>

typedef float v2f __attribute__((ext_vector_type(2)));
typedef float v8f __attribute__((ext_vector_type(8)));

#define VSZ  32128
#define HID  16
#define OUTD 16
#define SEQ  512
#define ROWS 16          // batch rows per block
#define TPR  16          // threads cooperating per row
#define BLK  (ROWS * TPR)

__device__ __forceinline__ float4 f4add(float4 a, float4 b) {
    return make_float4(a.x + b.x, a.y + b.y, a.z + b.z, a.w + b.w);
}

// enc1_w is [H=16, V] row-major; produce wT [V, 16] so each token's column is
// one contiguous 64-byte row (4x b128 loads per gather).
__global__ void transpose_w_kernel(const float* __restrict__ w,
                                   float* __restrict__ wT) {
    int v = blockIdx.x * blockDim.x + threadIdx.x;
    if (v >= VSZ) return;
#pragma unroll
    for (int k = 0; k < HID; ++k)
        wT[(size_t)v * HID + k] = w[(size_t)k * VSZ + v];
}

__global__ void __launch_bounds__(BLK)
encoder_kernel(const int* __restrict__ x,        // [B, SEQ]
               const float* __restrict__ wT,     // [V, 16]
               const float* __restrict__ enc1_b, // [16]
               const float* __restrict__ mean_w, // [16,16] row-major [O,H]
               const float* __restrict__ mean_b, // [16]
               const float* __restrict__ lv_w,   // [16,16]
               const float* __restrict__ lv_b,   // [16]
               float* __restrict__ out_mean,     // [B,16]
               float* __restrict__ out_lv,       // [B,16]
               int Btot) {
    __shared__ float part[ROWS][TPR][HID];  // 16 KB partial sums
    __shared__ float hmat[ROWS][HID];       // 1 KB activated hidden tile

    const int tid = threadIdx.x;
    const int row = tid >> 4;   // local batch row 0..15
    const int tr  = tid & 15;   // thread-in-row 0..15
    const int b   = blockIdx.x * ROWS + row;

    // ---- gather-sum phase: h_partial = sum over my tokens of wT[tok, :] ----
    float4 a0 = make_float4(0.f, 0.f, 0.f, 0.f);
    float4 a1 = a0, a2 = a0, a3 = a0;
    if (b < Btot) {
        const int* xrow = x + (size_t)b * SEQ;
        const float4* wT4 = (const float4*)wT;  // 4 float4 per token row
        for (int s = tr; s < SEQ; s += TPR) {
            int tok = xrow[s];                  // coalesced across 16 lanes
            const float4* wr = wT4 + (size_t)tok * 4;
            a0 = f4add(a0, wr[0]);
            a1 = f4add(a1, wr[1]);
            a2 = f4add(a2, wr[2]);
            a3 = f4add(a3, wr[3]);
        }
    }
    {
        float4* p = (float4*)&part[row][tr][0];
        p[0] = a0; p[1] = a1; p[2] = a2; p[3] = a3;
    }
    __syncthreads();

    // ---- reduce 16 partials per row, add bias, ReLU -> hmat[16][16] ----
    {
        const int r = row, k = tr;  // 256 threads = all (r,k) pairs
        float s = 0.f;
#pragma unroll
        for (int t = 0; t < TPR; ++t) s += part[r][t][k];
        s += enc1_b[k];
        hmat[r][k] = s > 0.f ? s : 0.f;
    }
    __syncthreads();

    // ---- projection phase: wave 0 only (EXEC all-ones in this wave) ----
    if (tid < 32) {
        const int lane = tid;
        const int half = lane >> 4;  // 0: lanes 0-15, 1: lanes 16-31
        const int l16  = lane & 15;

        // A = h (16x16), split along K into 4 chunks of 4.
        // 32-bit A 16x4 layout: lanes 0-15 hold M=lane {K0,K1}; lanes 16-31 {K2,K3}
        v2f a[4];
#pragma unroll
        for (int c = 0; c < 4; ++c) {
            int k0 = 4 * c + 2 * half;
            a[c].x = hmat[l16][k0];
            a[c].y = hmat[l16][k0 + 1];
        }
        // B[k][n] = W[n][k] (since out = h @ W^T). 32-bit B 4x16 layout:
        // VGPR0: K0 row (lanes 0-15) / K2 row (lanes 16-31); VGPR1: K1 / K3
        v2f bm[4], bl[4];
#pragma unroll
        for (int c = 0; c < 4; ++c) {
            int k0 = 4 * c + 2 * half;
            bm[c].x = mean_w[l16 * HID + k0];
            bm[c].y = mean_w[l16 * HID + k0 + 1];
            bl[c].x = lv_w[l16 * HID + k0];
            bl[c].y = lv_w[l16 * HID + k0 + 1];
        }

        v8f accm = {};
#pragma unroll
        for (int c = 0; c < 4; ++c)
            accm = __builtin_amdgcn_wmma_f32_16x16x4_f32(
                false, a[c], false, bm[c], (short)0, accm, false, false);
        v8f accl = {};
#pragma unroll
        for (int c = 0; c < 4; ++c)
            accl = __builtin_amdgcn_wmma_f32_16x16x4_f32(
                false, a[c], false, bl[c], (short)0, accl, false, false);

        // C/D layout: VGPR j -> M=j (lanes 0-15) / M=j+8 (lanes 16-31), N=l16
        const float mb = mean_b[l16];
        const float lb = lv_b[l16];
        const int b0 = blockIdx.x * ROWS;
#pragma unroll
        for (int j = 0; j < 8; ++j) {
            int m = j + 8 * half;
            int gb = b0 + m;
            if (gb < Btot) {
                out_mean[(size_t)gb * OUTD + l16] = accm[j] + mb;
                out_lv[(size_t)gb * OUTD + l16]   = accl[j] + lb;
            }
        }
    }
}

extern "C" void kernel_launch(void* const* d_in, const int* in_sizes, int n_in,
                              void* d_out, int out_size, void* d_ws, size_t ws_size,
                              hipStream_t stream) {
    const int*   x      = (const int*)d_in[0];
    const float* enc1_w = (const float*)d_in[1];
    const float* enc1_b = (const float*)d_in[2];
    const float* mean_w = (const float*)d_in[3];
    const float* mean_b = (const float*)d_in[4];
    const float* lv_w   = (const float*)d_in[5];
    const float* lv_b   = (const float*)d_in[6];

    const int Btot = in_sizes[0] / SEQ;  // 2048

    float* wT = (float*)d_ws;  // [VSZ, 16] = ~2 MB scratch

    transpose_w_kernel<<<(VSZ + 255) / 256, 256, 0, stream>>>(enc1_w, wT);

    float* out_mean = (float*)d_out;
    float* out_lv   = (float*)d_out + (size_t)Btot * OUTD;
    encoder_kernel<<<(Btot + ROWS - 1) / ROWS, BLK, 0, stream>>>(
        x, wT, enc1_b, mean_w, mean_b, lv_w, lv_b, out_mean, out_lv, Btot);
}